// QuantumFeedForward_65481071406372
// MI455X (gfx1250) — compile-verified
//
#include <hip/hip_runtime.h>
#include <math.h>

typedef __attribute__((ext_vector_type(2))) float v2f;
typedef __attribute__((ext_vector_type(8))) float v8f;

// ---------------------------------------------------------------------------
// Stage 1: q = circuit_expvals(params); v[g] = relu(dot(q, w2[g,:])), g<4096
// One block, trivial cost. w2 is (4096,4) row-major -> float4 per row.
// ---------------------------------------------------------------------------
__global__ void qff_stage1(const float* __restrict__ w2,
                           const float* __restrict__ params,
                           float* __restrict__ v) {
  __shared__ float qs[4];
  if (threadIdx.x == 0) {
    float c0 = cosf(params[0]);
    float c1 = cosf(params[1]);
    float c2 = cosf(params[2]);
    float c3 = cosf(params[3]);
    qs[0] = c1 * c2 * c3;
    qs[1] = c0 * c1;
    qs[2] = c0 * c1 * c2;
    qs[3] = c0 * c1 * c2 * c3;
  }
  __syncthreads();
  const float q0 = qs[0], q1 = qs[1], q2 = qs[2], q3 = qs[3];
#pragma unroll
  for (int i = 0; i < 16; ++i) {
    const int g = threadIdx.x + i * 256;
    const float4 w = ((const float4*)w2)[g];
    float h = q0 * w.x + q1 * w.y + q2 * w.z + q3 * w.w;
    v[g] = h > 0.0f ? h : 0.0f;
  }
}

// ---------------------------------------------------------------------------
// Stage 2: out_vec[e] = sum_g v[g] * w_out[e,g]  via V_WMMA_F32_16X16X4_F32.
// One wave per 16-element chunk of e (64 waves total: 16 blocks x 4 waves).
// A (16x4, M=e rows): lanes 0-15 -> M=lane, K={0,1}; lanes 16-31 -> K={2,3}.
// B (4x16): every column n gets v[g+k]; per-lane we supply the two K slots
// for this lane's half (k0 = 0 or 2), value independent of n.
// All 16 columns of D are identical; lanes 0 and 16 hold M=0..7 / M=8..15.
// w_out (1024,4096) row-major is streamed once with b64 loads; v from LDS.
// ---------------------------------------------------------------------------
__global__ void qff_stage2(const float* __restrict__ w_out,
                           const float* __restrict__ v,
                           float* __restrict__ out_vec) {
  __shared__ __align__(16) float vs[4096];
  {
    const float4* src = (const float4*)v;
    float4* dst = (float4*)vs;
    for (int i = threadIdx.x; i < 1024; i += blockDim.x) dst[i] = src[i];
  }
  __syncthreads();

  const int wave = blockIdx.x * (blockDim.x >> 5) + (threadIdx.x >> 5);
  const int lane = threadIdx.x & 31;
  const int m    = lane & 15;
  const int k0   = (lane >> 4) * 2;       // 0 for lanes 0-15, 2 for lanes 16-31
  const int e0   = wave * 16;

  const float* arow = w_out + (size_t)(e0 + m) * 4096 + k0;  // 8B aligned
  const float* brow = vs + k0;                               // 8B aligned

  v8f acc = {};
  for (int g = 0; g < 4096; g += 4) {
    const v2f a = *(const v2f*)(arow + g);   // A: w_out[e0+m][g+k0 .. +1]
    const v2f b = *(const v2f*)(brow + g);   // B: v[g+k0 .. +1] (all columns)
    acc = __builtin_amdgcn_wmma_f32_16x16x4_f32(
        /*neg_a=*/false, a, /*neg_b=*/false, b,
        /*c_mod=*/(short)0, acc, /*reuse_a=*/false, /*reuse_b=*/false);
  }

  if (lane == 0) {
#pragma unroll
    for (int r = 0; r < 8; ++r) out_vec[e0 + r] = acc[r];
  } else if (lane == 16) {
#pragma unroll
    for (int r = 0; r < 8; ++r) out_vec[e0 + 8 + r] = acc[r];
  }
}

// ---------------------------------------------------------------------------
// Stage 3: broadcast out_vec (1024 f32) to all B*S rows. Pure b128 stores,
// HBM-bandwidth bound (64 MiB). Each block writes 8 rows; thread t owns
// columns [4t, 4t+4) and reuses its cached float4.
// ---------------------------------------------------------------------------
__global__ void qff_stage3(const float* __restrict__ out_vec,
                           float* __restrict__ out) {
  const float4 ov = ((const float4*)out_vec)[threadIdx.x];
  float4* o = (float4*)(out + (size_t)blockIdx.x * 8 * 1024) + threadIdx.x;
#pragma unroll
  for (int i = 0; i < 8; ++i) {
    o[i * 256] = ov;   // row stride = 1024 floats = 256 float4
  }
}

extern "C" void kernel_launch(void* const* d_in, const int* in_sizes, int n_in,
                              void* d_out, int out_size, void* d_ws, size_t ws_size,
                              hipStream_t stream) {
  // Inputs (setup_inputs order): x, w1, w2, w_out, params.
  // x and w1 feed a dead path (RZ on |0> is a pure phase) -> unused.
  (void)n_in; (void)ws_size;
  const float* w2     = (const float*)d_in[2];   // (4096, 4)
  const float* w_out  = (const float*)d_in[3];   // (1024, 4096)
  const float* params = (const float*)d_in[4];   // (4,)

  float* v       = (float*)d_ws;     // 4096 floats
  float* out_vec = v + 4096;         // 1024 floats

  qff_stage1<<<1, 256, 0, stream>>>(w2, params, v);
  qff_stage2<<<16, 128, 0, stream>>>(w_out, v, out_vec);

  const int rows = out_size / 1024;              // B*S = 16384
  qff_stage3<<<rows / 8, 256, 0, stream>>>(out_vec, (float*)d_out);
}